// ViewMorphing_70145405878810
// MI455X (gfx1250) — compile-verified
//
#include <hip/hip_runtime.h>

// ViewMorphing warp: out[n,c,p] = im1[n,c,clip(b+d)]*M1[n,p] + im2[n,c,clip(b-d)]*M2[n,p]
// b = (p/224) + 224*(p%224)  (transposed index), d = C0 + 224*C1, C in [0,5).
// Memory-bound (~334 MB compulsory -> ~14 us @ 23.3 TB/s). Zero matmul FLOPs -> no WMMA.
// CDNA5 data-movement paths used instead:
//   * GLOBAL_LOAD_ASYNC_TO_LDS_B32  : coalesced load + per-lane LDS scatter (transpose in
//     the async unit, no VGPR staging), ASYNCcnt-tracked, NT hint.
//   * GLOBAL_STORE_ASYNC_FROM_LDS_B32: per-lane LDS gather (transpose) + coalesced store.
//   * NT temporal hints keep the 192 MB L2 free for the 154 MB im1+im2 gather set.

#define IMAGE_DIM 224
#define HW        (IMAGE_DIM * IMAGE_DIM)   // 50176
#define TILE      32
#define TPB       256
#define PITCH     36                        // LDS pitch: 16B alignment, spreads banks

typedef __attribute__((ext_vector_type(4))) float v4f;
typedef __attribute__((ext_vector_type(4))) int   v4i;

// Generic pointer to LDS: shared aperture => addr[31:0] is the LDS byte offset.
__device__ __forceinline__ unsigned lds_off(const void* p) {
    return (unsigned)(unsigned long long)p;
}

// Async global -> LDS, 4B per lane; lane's LDS destination address in a VGPR.
__device__ __forceinline__ void async_g2l_b32(unsigned ldsoff, const void* gaddr) {
    asm volatile("global_load_async_to_lds_b32 %0, %1, off th:TH_LOAD_NT"
                 :: "v"(ldsoff), "v"((unsigned long long)gaddr)
                 : "memory");
}

// Async LDS -> global, 4B per lane; lane's LDS source address in a VGPR.
__device__ __forceinline__ void async_l2g_b32(const void* gaddr, unsigned ldsoff) {
    asm volatile("global_store_async_from_lds_b32 %0, %1, off th:TH_STORE_NT"
                 :: "v"((unsigned long long)gaddr), "v"(ldsoff)
                 : "memory");
}

__device__ __forceinline__ void wait_async0() {
    asm volatile("s_wait_asynccnt 0x0" ::: "memory");
}

__global__ __launch_bounds__(TPB) void viewmorph_kernel(
    const float* __restrict__ im1, const float* __restrict__ im2,
    const int*   __restrict__ C,   const float* __restrict__ M1,
    const float* __restrict__ M2,  float* __restrict__ out)
{
    __shared__ __align__(16) int   sC0[TILE * PITCH];
    __shared__ __align__(16) int   sC1[TILE * PITCH];
    __shared__ __align__(16) float sM1[TILE * PITCH];
    __shared__ __align__(16) float sM2[TILE * PITCH];
    __shared__ __align__(16) float sO[3][TILE * PITCH];

    const int bid  = blockIdx.x;
    const int n    = bid / 49;
    const int tile = bid % 49;
    const int ty0  = (tile / 7) * TILE;     // tile origin in y (rows of p)
    const int tx0  = (tile % 7) * TILE;     // tile origin in x (cols of p)
    const int t    = threadIdx.x;

    // ---- Phase 1: async global->LDS with transposed per-lane LDS destinations.
    // Lanes consecutive in x => each op is a fully coalesced 128B/wave global read.
    {
        const int cb = n * 2 * HW;
        const int mb = n * HW;
        #pragma unroll
        for (int j = 0; j < 4; ++j) {
            const int e  = t + j * TPB;                     // 0..1023 tile pixel
            const int r  = e >> 5;                          // tile-local y
            const int xq = e & 31;                          // tile-local x
            const int gp = (ty0 + r) * IMAGE_DIM + tx0 + xq;
            const int li = xq * PITCH + r;                  // LDS [x][y]
            async_g2l_b32(lds_off(&sC0[li]), C  + cb      + gp);
            async_g2l_b32(lds_off(&sC1[li]), C  + cb + HW + gp);
            async_g2l_b32(lds_off(&sM1[li]), M1 + mb      + gp);
            async_g2l_b32(lds_off(&sM2[li]), M2 + mb      + gp);
        }
    }
    wait_async0();
    __syncthreads();

    // ---- Phase 2: lanes contiguous in y => gather addresses cluster into ~5 runs.
    {
        const int xl    = t >> 3;           // tile-local x, 0..31
        const int y4    = (t & 7) << 2;     // tile-local y quad
        const int lbase = xl * PITCH + y4;  // 16B aligned (PITCH, y4 mult of 4)
        const v4i c0 = *(const v4i*)&sC0[lbase];
        const v4i c1 = *(const v4i*)&sC1[lbase];
        const v4f m1 = *(const v4f*)&sM1[lbase];
        const v4f m2 = *(const v4f*)&sM2[lbase];
        const float* im1n = im1 + n * 3 * HW;
        const float* im2n = im2 + n * 3 * HW;
        const int x = tx0 + xl;
        v4f o0, o1, o2;
        #pragma unroll
        for (int i = 0; i < 4; ++i) {
            const int y    = ty0 + y4 + i;
            const int base = y + IMAGE_DIM * x;          // transposed gather base
            const int d    = c0[i] + IMAGE_DIM * c1[i];  // 0..900
            int i1 = min(max(base + d, 0), HW - 1);      // mode='clip'
            int i2 = min(max(base - d, 0), HW - 1);
            o0[i] = fmaf(im1n[         i1], m1[i], im2n[         i2] * m2[i]);
            o1[i] = fmaf(im1n[HW     + i1], m1[i], im2n[HW     + i2] * m2[i]);
            o2[i] = fmaf(im1n[2 * HW + i1], m1[i], im2n[2 * HW + i2] * m2[i]);
        }
        *(v4f*)&sO[0][lbase] = o0;
        *(v4f*)&sO[1][lbase] = o1;
        *(v4f*)&sO[2][lbase] = o2;
    }
    __syncthreads();

    // ---- Phase 3: async LDS->global; per-lane transposed LDS reads, coalesced
    // 128B/wave global writes. S_ENDPGM's implicit wait-idle drains ASYNCcnt.
    {
        float* outn = out + n * 3 * HW;
        #pragma unroll
        for (int j = 0; j < 4; ++j) {
            const int e  = t + j * TPB;
            const int r  = e >> 5;
            const int xq = e & 31;
            const int li = xq * PITCH + r;
            float* g = outn + (ty0 + r) * IMAGE_DIM + tx0 + xq;
            #pragma unroll
            for (int c = 0; c < 3; ++c)
                async_l2g_b32(g + c * HW, lds_off(&sO[c][li]));
        }
    }
}

extern "C" void kernel_launch(void* const* d_in, const int* in_sizes, int n_in,
                              void* d_out, int out_size, void* d_ws, size_t ws_size,
                              hipStream_t stream) {
    const float* im1 = (const float*)d_in[0];
    const float* im2 = (const float*)d_in[1];
    const int*   C   = (const int*)  d_in[2];
    const float* M1  = (const float*)d_in[3];
    const float* M2  = (const float*)d_in[4];
    float*       out = (float*)d_out;

    const int N = in_sizes[0] / (3 * HW);   // 128 for the reference setup
    // 224 = 7*32 exactly -> 49 full tiles per image, no edge handling needed.
    viewmorph_kernel<<<dim3(N * 49), TPB, 0, stream>>>(im1, im2, C, M1, M2, out);
}